// GraphSage_3118146256900
// MI455X (gfx1250) — compile-verified
//
#include <hip/hip_runtime.h>

typedef float v2f __attribute__((ext_vector_type(2)));
typedef float v8f __attribute__((ext_vector_type(8)));

#define D_IN   256
#define D_OUT  256
#define BATCH  4096
#define K1     25
#define K2     10
#define MROWS  (BATCH * (1 + K2))   // 45056
#define DCAT   512                  // 2*D_IN == 2*D_OUT

// ---------------------------------------------------------------------------
// Kernel 1: layer-1 gather.  X1[m] = [ feat[l1_nodes[m]] || mean_j feat[neigh1[m,j]] ]
// 256 threads = 4 groups of 64; each group builds one row with float4 (b128) loads.
// ---------------------------------------------------------------------------
__global__ __launch_bounds__(256) void sage_gather1(
    const float* __restrict__ feat,        // [N_NODES, 256]
    const int*   __restrict__ nodes_batch, // [BATCH]
    const int*   __restrict__ neigh2,      // [BATCH*K2] flattened
    const int*   __restrict__ neigh1,      // [MROWS*K1]
    float*       __restrict__ X1)          // [MROWS, 512]
{
    const int grp = threadIdx.x >> 6;
    const int t   = threadIdx.x & 63;
    const int m   = blockIdx.x * 4 + grp;  // grid is exactly MROWS/4
    const int d   = t << 2;

    const int self_id = (m < BATCH) ? nodes_batch[m] : neigh2[m - BATCH];
    const float4 s = *(const float4*)(feat + (size_t)self_id * D_IN + d);

    const int* nrow = neigh1 + (size_t)m * K1;
    float ax = 0.f, ay = 0.f, az = 0.f, aw = 0.f;
    int id = nrow[0];
    #pragma unroll 5
    for (int j = 0; j < K1; ++j) {
        const float4 v = *(const float4*)(feat + (size_t)id * D_IN + d);
        if (j + 1 < K1) {
            id = nrow[j + 1];
            __builtin_prefetch(feat + (size_t)id * D_IN + d, 0, 1); // global_prefetch_b8
        }
        ax += v.x; ay += v.y; az += v.z; aw += v.w;
    }
    const float inv = 1.0f / (float)K1;

    float* out = X1 + (size_t)m * DCAT + d;
    *(float4*)out = s;
    *(float4*)(out + D_IN) = make_float4(ax * inv, ay * inv, az * inv, aw * inv);
}

// ---------------------------------------------------------------------------
// Kernels 2 & 4: H = relu(X @ W^T) via V_WMMA_F32_16X16X4_F32.
// X: [rows, 512] row-major, W: [256, 512] row-major (so B = W^T needs no transpose).
// One wave -> 16x64 output strip; 4 waves/block -> full 256 columns of one M-strip.
// ---------------------------------------------------------------------------
__global__ __launch_bounds__(128) void sage_gemm_relu_wmma(
    const float* __restrict__ X,   // [rows, 512]
    const float* __restrict__ W,   // [256, 512]
    float*       __restrict__ H)   // [rows, 256]
{
    const int lane = threadIdx.x & 31;
    const int wave = threadIdx.x >> 5;   // 0..3 -> 64-column slab
    const int half = lane >> 4;          // K sub-pair selector per ISA A/B layout
    const int l16  = lane & 15;

    const size_t m0 = (size_t)blockIdx.x * 16;
    const int    n0 = wave * 64;

    const float* xrow = X + (m0 + l16) * DCAT;
    const float* wr0  = W + (size_t)(n0 +  0 + l16) * DCAT;
    const float* wr1  = W + (size_t)(n0 + 16 + l16) * DCAT;
    const float* wr2  = W + (size_t)(n0 + 32 + l16) * DCAT;
    const float* wr3  = W + (size_t)(n0 + 48 + l16) * DCAT;

    v8f acc0 = {}, acc1 = {}, acc2 = {}, acc3 = {};

    #pragma unroll 8
    for (int k = 0; k < DCAT; k += 4) {
        const int kk = k + 2 * half;
        const v2f a  = *(const v2f*)(xrow + kk);
        const v2f b0 = *(const v2f*)(wr0 + kk);
        const v2f b1 = *(const v2f*)(wr1 + kk);
        const v2f b2 = *(const v2f*)(wr2 + kk);
        const v2f b3 = *(const v2f*)(wr3 + kk);
        acc0 = __builtin_amdgcn_wmma_f32_16x16x4_f32(false, a, false, b0, (short)0, acc0, false, false);
        acc1 = __builtin_amdgcn_wmma_f32_16x16x4_f32(false, a, false, b1, (short)0, acc1, false, false);
        acc2 = __builtin_amdgcn_wmma_f32_16x16x4_f32(false, a, false, b2, (short)0, acc2, false, false);
        acc3 = __builtin_amdgcn_wmma_f32_16x16x4_f32(false, a, false, b3, (short)0, acc3, false, false);
    }

    // D layout: VGPR r -> row (r + 8*half), lane -> column; fuse ReLU.
    #pragma unroll
    for (int r = 0; r < 8; ++r) {
        float* hrow = H + (m0 + r + 8 * half) * D_OUT + n0 + l16;
        hrow[0]  = fmaxf(acc0[r], 0.0f);
        hrow[16] = fmaxf(acc1[r], 0.0f);
        hrow[32] = fmaxf(acc2[r], 0.0f);
        hrow[48] = fmaxf(acc3[r], 0.0f);
    }
}

// ---------------------------------------------------------------------------
// Kernel 3: X2[b] = [ h1[b] || mean_k h1[BATCH + b*K2 + k] ]
// ---------------------------------------------------------------------------
__global__ __launch_bounds__(256) void sage_build_x2(
    const float* __restrict__ h1,   // [MROWS, 256]
    float*       __restrict__ X2)   // [BATCH, 512]
{
    const int grp = threadIdx.x >> 6;
    const int t   = threadIdx.x & 63;
    const int b   = blockIdx.x * 4 + grp;  // grid is exactly BATCH/4
    const int d   = t << 2;

    const float4 s = *(const float4*)(h1 + (size_t)b * D_OUT + d);

    const float* base = h1 + ((size_t)BATCH + (size_t)b * K2) * D_OUT + d;
    float ax = 0.f, ay = 0.f, az = 0.f, aw = 0.f;
    #pragma unroll
    for (int j = 0; j < K2; ++j) {
        const float4 v = *(const float4*)(base + (size_t)j * D_OUT);
        ax += v.x; ay += v.y; az += v.z; aw += v.w;
    }
    const float inv = 1.0f / (float)K2;

    float* out = X2 + (size_t)b * DCAT + d;
    *(float4*)out = s;
    *(float4*)(out + D_OUT) = make_float4(ax * inv, ay * inv, az * inv, aw * inv);
}

// ---------------------------------------------------------------------------
extern "C" void kernel_launch(void* const* d_in, const int* in_sizes, int n_in,
                              void* d_out, int out_size, void* d_ws, size_t ws_size,
                              hipStream_t stream) {
    (void)in_sizes; (void)n_in; (void)out_size; (void)ws_size;

    const float* feat        = (const float*)d_in[0];
    const float* W1          = (const float*)d_in[1];
    const float* W2          = (const float*)d_in[2];
    const int*   nodes_batch = (const int*)d_in[3];
    const int*   neigh2      = (const int*)d_in[4];
    const int*   neigh1      = (const int*)d_in[5];

    float* X1 = (float*)d_ws;                        // [45056, 512]  ~92 MB
    float* h1 = X1 + (size_t)MROWS * DCAT;           // [45056, 256]  ~46 MB
    float* X2 = h1 + (size_t)MROWS * D_OUT;          // [ 4096, 512]  ~8 MB
    float* out = (float*)d_out;                      // [ 4096, 256]

    sage_gather1      <<<MROWS / 4, 256, 0, stream>>>(feat, nodes_batch, neigh2, neigh1, X1);
    sage_gemm_relu_wmma<<<MROWS / 16, 128, 0, stream>>>(X1, W1, h1);
    sage_build_x2     <<<BATCH / 4, 256, 0, stream>>>(h1, X2);
    sage_gemm_relu_wmma<<<BATCH / 16, 128, 0, stream>>>(X2, W2, out);
}